// MultiInnerProductDecoder2_15367392985217
// MI455X (gfx1250) — compile-verified
//
#include <hip/hip_runtime.h>

typedef __attribute__((ext_vector_type(2))) float v2f;
typedef __attribute__((ext_vector_type(8))) float v8f;

#define N_NODES      100000
#define IN_DIM       128
#define NUM_ET       16
#define E_PER_TYPE   100000
#define GROUPS_PER_TYPE (E_PER_TYPE / 16)   // 6250 groups of 16 edges

// One wave handles 16 edges per group iteration:
//   A (16x4 f32 per chunk): A[m][k] = z[src_m][4c+k]
//   B (4x16 f32 per chunk): B[k][n] = z[dst_n][4c+k] * w[4c+k]
//   D += A x B  via V_WMMA_F32_16X16X4_F32, 32 chunks over IN_DIM=128
//   diag(D)[e]  = <z[src_e], z[dst_e] * w>   (off-diagonal is discarded)
//
// f32 A-matrix 16x4 VGPR layout (ISA 7.12.2): lane L<16 holds {A[L][K],A[L][K+1]},
// lane L+16 holds {A[L][K+2],A[L][K+3]}  -> one global_load_b64 per lane per chunk
// with base = row + 2*(lane>>4), immediate offset = 16*c bytes.
// B assumed to mirror A's half-wave K split (K / K+2 in the register pair).
__global__ __launch_bounds__(256)
void MultiInnerProductDecoder2_kernel(const float* __restrict__ z,
                                      const float* __restrict__ weight,
                                      const long long* __restrict__ edge_index,
                                      float* __restrict__ out)
{
    const int t    = blockIdx.y;
    const int lane = threadIdx.x & 31;
    const int wavesPerBlock = blockDim.x >> 5;
    const int wid  = blockIdx.x * wavesPerBlock + (threadIdx.x >> 5);
    const int wavesTotal = gridDim.x * wavesPerBlock;

    const int m    = lane & 15;        // row (for A) / col (for B) index 0..15
    const int koff = (lane >> 4) << 1; // 0 for lanes 0-15, 2 for lanes 16-31

    // Per-lane weight pairs are loop-invariant for this block's edge type:
    // wv[c] = weight[t][4c + koff .. 4c + koff + 1]   (64 VGPRs, reused forever)
    const float* wrow = weight + (size_t)t * IN_DIM;
    v2f wv[32];
#pragma unroll
    for (int c = 0; c < 32; ++c)
        wv[c] = *(const v2f*)(wrow + c * 4 + koff);

    const long long* ei_src = edge_index + (long long)t * (2LL * E_PER_TYPE);
    const long long* ei_dst = ei_src + E_PER_TYPE;

    for (int g = wid; g < GROUPS_PER_TYPE; g += wavesTotal) {
        const int e0 = g * 16;

        // Both half-waves read the same 16 src/dst indices (broadcast, L2/L0 hit)
        const long long s = ei_src[e0 + m];
        const long long d = ei_dst[e0 + m];

        const float* abase = z + (long long)s * IN_DIM + koff; // 8B aligned
        const float* bbase = z + (long long)d * IN_DIM + koff;

        v8f acc = {};
#pragma unroll
        for (int c = 0; c < 32; ++c) {
            v2f a = *(const v2f*)(abase + c * 4);  // global_load_b64, imm offset 16*c
            v2f b = *(const v2f*)(bbase + c * 4);
            b *= wv[c];                            // fold weight into B
            acc = __builtin_amdgcn_wmma_f32_16x16x4_f32(
                /*neg_a=*/false, a, /*neg_b=*/false, b,
                /*c_mod=*/(short)0, acc, /*reuse_a=*/false, /*reuse_b=*/false);
        }

        // Diagonal extraction. C/D layout: D[M][N] at vgpr=M%8,
        // lane = N + 16*(M/8). So D[e][e] sits at:
        //   e<8 : lane e,     vgpr e
        //   e>=8: lane e+16,  vgpr e-8      (i.e. lanes 24..31)
        const int idx = ((lane < 16) ? lane : (lane - 24)) & 7;
        float p0 = (idx & 1) ? acc[1] : acc[0];
        float p1 = (idx & 1) ? acc[3] : acc[2];
        float p2 = (idx & 1) ? acc[5] : acc[4];
        float p3 = (idx & 1) ? acc[7] : acc[6];
        float q0 = (idx & 2) ? p1 : p0;
        float q1 = (idx & 2) ? p3 : p2;
        float v  = (idx & 4) ? q1 : q0;

        // sigmoid(v) = 1 / (1 + exp(-v))  (v_exp_f32 + v_rcp_f32 path)
        float sgm = __builtin_amdgcn_rcpf(1.0f + __expf(-v));

        const bool valid = (lane < 8) | (lane >= 24);
        if (valid) {
            const int e = (lane < 8) ? lane : (lane - 16);   // 0..15
            const long long o = (long long)t * E_PER_TYPE + e0 + e;
            out[o] = sgm;                                    // per_type
            out[o + (long long)NUM_ET * E_PER_TYPE] = sgm;   // score (same values)
        }
    }
}

extern "C" void kernel_launch(void* const* d_in, const int* in_sizes, int n_in,
                              void* d_out, int out_size, void* d_ws, size_t ws_size,
                              hipStream_t stream) {
    const float*     z  = (const float*)d_in[0];      // (100000,128) f32
    const float*     w  = (const float*)d_in[1];      // (16,128)     f32
    const long long* ei = (const long long*)d_in[2];  // (16,2,100000) i64
    float* out = (float*)d_out;                       // 1.6M per_type + 1.6M score

    dim3 grid(98, NUM_ET, 1);   // 784 waves per type, ~8 groups each
    dim3 block(256, 1, 1);      // 8 waves/block (wave32)
    MultiInnerProductDecoder2_kernel<<<grid, block, 0, stream>>>(z, w, ei, out);
}